// RotatE_72576357368235
// MI455X (gfx1250) — compile-verified
//
#include <hip/hip_runtime.h>
#include <math.h>
#include <stdint.h>

#define EMB_DIM       500
#define NUM_ENTITIES  14541
#define BQ            16
#define GAMMA_F       12.0f
// phase scale = PI / ((GAMMA + 2) / EMB_DIM)
#define PHASE_SCALE   (3.14159265358979323846f / ((12.0f + 2.0f) / 500.0f))
#define ROT_ELEMS     (2 * BQ * EMB_DIM)   /* 16000 floats = 64 KB */

// ---------------------------------------------------------------------------
// Kernel 1: gather + complex-rotate the 16 head embeddings into d_ws,
// INTERLEAVED as (re,im) pairs:  rot[2*(b*500+d)] = re_rot, +1 = im_rot.
// ---------------------------------------------------------------------------
__global__ void rotate_phase_kernel(const int* __restrict__ all_h,
                                    const int* __restrict__ all_r,
                                    const float* __restrict__ eemb,
                                    const float* __restrict__ remb,
                                    float* __restrict__ rot) {
  int idx = blockIdx.x * blockDim.x + threadIdx.x;
  if (idx >= BQ * EMB_DIM) return;
  int b = idx / EMB_DIM;
  int d = idx - b * EMB_DIM;
  int h = all_h[b];
  int r = all_r[b];
  const float* hrow = eemb + (size_t)h * (2 * EMB_DIM);
  float re_h = hrow[d];
  float im_h = hrow[EMB_DIM + d];
  float phase = remb[(size_t)r * EMB_DIM + d] * PHASE_SCALE;
  float s, c;
  __sincosf(phase, &s, &c);
  float2 v;
  v.x = re_h * c - im_h * s;   // re_rot
  v.y = re_h * s + im_h * c;   // im_rot
  *(float2*)(rot + 2 * idx) = v;   // 8B-aligned interleaved store
}

// ---------------------------------------------------------------------------
// Folding butterfly: reduce 16 per-lane accumulators across 32 lanes.
// 16 shuffles + 16 adds per call. On return, lane l holds the total for
// batch b = l>>1 (lanes 2b and 2b+1 hold identical sums).
// ---------------------------------------------------------------------------
__device__ __forceinline__ float reduce16x32(float* a, int lane) {
#pragma unroll
  for (int k = 0; k < 8; ++k) {
    bool hi = (lane & 16) != 0;
    float keep = hi ? a[k + 8] : a[k];
    float send = hi ? a[k] : a[k + 8];
    a[k] = keep + __shfl_xor(send, 16, 32);
  }
#pragma unroll
  for (int k = 0; k < 4; ++k) {
    bool hi = (lane & 8) != 0;
    float keep = hi ? a[k + 4] : a[k];
    float send = hi ? a[k] : a[k + 4];
    a[k] = keep + __shfl_xor(send, 8, 32);
  }
#pragma unroll
  for (int k = 0; k < 2; ++k) {
    bool hi = (lane & 4) != 0;
    float keep = hi ? a[k + 2] : a[k];
    float send = hi ? a[k] : a[k + 2];
    a[k] = keep + __shfl_xor(send, 4, 32);
  }
  {
    bool hi = (lane & 2) != 0;
    float keep = hi ? a[1] : a[0];
    float send = hi ? a[0] : a[1];
    a[0] = keep + __shfl_xor(send, 2, 32);
  }
  a[0] += __shfl_xor(a[0], 1, 32);
  return a[0];
}

// ---------------------------------------------------------------------------
// Kernel 2: one wave32 per ENTITY PAIR; lanes stride the D dimension two
// values at a time; rot table staged to LDS via async-to-LDS b128 copies and
// read back as float4 (re0,im0,re1,im1) -> ds_load_b128, reused for both
// entities. 16 ds + 4 vmem feed 320 VALU + 64 v_sqrt per step.
// ---------------------------------------------------------------------------
#define SCORE_PAIR(DBASE)                                                     \
  {                                                                           \
    const int d_ = (DBASE);                                                   \
    float2 ret0 = *(const float2*)(row0 + d_);                                \
    float2 imt0 = *(const float2*)(row0 + EMB_DIM + d_);                      \
    float2 ret1 = *(const float2*)(row1 + d_);                                \
    float2 imt1 = *(const float2*)(row1 + EMB_DIM + d_);                      \
    _Pragma("unroll")                                                         \
    for (int b = 0; b < BQ; ++b) {                                            \
      float4 rt = *(const float4*)(&s_rot[b * (2 * EMB_DIM) + 2 * d_]);       \
      float rd0 = rt.x - ret0.x, id0 = rt.y - imt0.x;                         \
      acc0[b] += __builtin_amdgcn_sqrtf(fmaf(id0, id0, rd0 * rd0));           \
      float rd1 = rt.z - ret0.y, id1 = rt.w - imt0.y;                         \
      acc0[b] += __builtin_amdgcn_sqrtf(fmaf(id1, id1, rd1 * rd1));           \
      float rd2 = rt.x - ret1.x, id2 = rt.y - imt1.x;                         \
      acc1[b] += __builtin_amdgcn_sqrtf(fmaf(id2, id2, rd2 * rd2));           \
      float rd3 = rt.z - ret1.y, id3 = rt.w - imt1.y;                         \
      acc1[b] += __builtin_amdgcn_sqrtf(fmaf(id3, id3, rd3 * rd3));           \
    }                                                                         \
  }

__launch_bounds__(256)
__global__ void rotate_score_kernel(const float* __restrict__ eemb,
                                    const float* __restrict__ rot,
                                    float* __restrict__ out) {
  __shared__ float s_rot[ROT_ELEMS];
  const int tid = threadIdx.x;

  // --- Stage 64 KB rotation table into LDS via async-to-LDS (gfx1250) ------
  for (int c = tid; c < ROT_ELEMS / 4; c += 256) {
    unsigned           lds_off = (unsigned)(uintptr_t)(&s_rot[c * 4]);
    unsigned long long ga      = (unsigned long long)(uintptr_t)(rot + c * 4);
    asm volatile("global_load_async_to_lds_b128 %0, %1, off"
                 :: "v"(lds_off), "v"(ga) : "memory");
  }
  asm volatile("s_wait_asynccnt 0" ::: "memory");
  __syncthreads();

  const int lane  = tid & 31;
  const int gwave = (blockIdx.x * 256 + tid) >> 5;
  const int nwave = gridDim.x * 8;
  const int dl    = 2 * lane;           // this lane's base d within a 64-chunk

  for (int e0 = 2 * gwave; e0 < NUM_ENTITIES; e0 += 2 * nwave) {
    const int  e1  = e0 + 1;
    const bool ok1 = (e1 < NUM_ENTITIES);
    const float* row0 = eemb + (size_t)e0 * (2 * EMB_DIM);
    const float* row1 = ok1 ? (eemb + (size_t)e1 * (2 * EMB_DIM)) : row0;

    // Prefetch both rows of the next pair (global_prefetch_b8).
    int ep = e0 + 2 * nwave;
    if (ep < NUM_ENTITIES) {
      __builtin_prefetch((const void*)((const char*)(eemb + (size_t)ep * (2 * EMB_DIM)) + lane * 128), 0, 3);
      if (ep + 1 < NUM_ENTITIES)
        __builtin_prefetch((const void*)((const char*)(eemb + (size_t)(ep + 1) * (2 * EMB_DIM)) + lane * 128), 0, 3);
    }

    float acc0[BQ], acc1[BQ];
#pragma unroll
    for (int b = 0; b < BQ; ++b) { acc0[b] = 0.0f; acc1[b] = 0.0f; }

    // d = 2*lane + 64*i ; i=0..6 always in range (max d+1 = 447)
    for (int i = 0; i < 7; ++i) {
      SCORE_PAIR(dl + (i << 6))
    }
    // tail: d in [448, 500) -> 26 pairs, lanes 0..25
    if (dl < 52) {
      SCORE_PAIR(dl + 448)
    }

    // Folding butterfly reductions (16 shuffles each). Lane l holds batch
    // b = l>>1; even lanes store entity0, odd lanes store entity1.
    float r0 = reduce16x32(acc0, lane);
    float r1 = reduce16x32(acc1, lane);
    const int b = lane >> 1;
    if ((lane & 1) == 0) {
      out[(size_t)b * NUM_ENTITIES + e0] = GAMMA_F - r0;
    } else if (ok1) {
      out[(size_t)b * NUM_ENTITIES + e1] = GAMMA_F - r1;
    }
  }
}

// ---------------------------------------------------------------------------
extern "C" void kernel_launch(void* const* d_in, const int* in_sizes, int n_in,
                              void* d_out, int out_size, void* d_ws, size_t ws_size,
                              hipStream_t stream) {
  const int*   all_h = (const int*)d_in[0];
  const int*   all_r = (const int*)d_in[1];
  const float* eemb  = (const float*)d_in[2];
  const float* remb  = (const float*)d_in[3];
  float*       out   = (float*)d_out;
  float*       rot   = (float*)d_ws;      // 16000 floats = 64 KB scratch

  (void)in_sizes; (void)n_in; (void)out_size; (void)ws_size;

  // Kernel 1: 8000 work-items.
  rotate_phase_kernel<<<(BQ * EMB_DIM + 255) / 256, 256, 0, stream>>>(
      all_h, all_r, eemb, remb, rot);

  // Kernel 2: 456 blocks * 8 wave32 = 3648 waves, ~2 entity-pairs per wave.
  rotate_score_kernel<<<456, 256, 0, stream>>>(eemb, rot, out);
}